// GCN2_pr_21053929685346
// MI455X (gfx1250) — compile-verified
//
#include <hip/hip_runtime.h>
#include <hip/hip_bf16.h>

#define IN_D 128
#define HID  64
#define DOC_D 256
#define OUT_D 16
#define LN_EPS 1e-5f

typedef __attribute__((ext_vector_type(2))) float v2f;
typedef __attribute__((ext_vector_type(8))) float v8f;

// ---------------------------------------------------------------------------
// 0) Workspace init: deg=1 (self loop), accum/sums/cnt = 0
// ---------------------------------------------------------------------------
__global__ void init_ws(float* __restrict__ deg, float* __restrict__ accum,
                        float* __restrict__ sums, float* __restrict__ cnt,
                        int n, long long accumN, int g) {
    long long i = (long long)blockIdx.x * blockDim.x + threadIdx.x;
    long long stride = (long long)gridDim.x * blockDim.x;
    for (long long j = i; j < accumN; j += stride) accum[j] = 0.0f;
    for (long long j = i; j < n; j += stride) deg[j] = 1.0f;
    if (i < (long long)g * HID) sums[i] = 0.0f;
    if (i < g) cnt[i] = 0.0f;
}

// ---------------------------------------------------------------------------
// 1) Degree accumulation over edge destinations
// ---------------------------------------------------------------------------
__global__ void deg_accum(const long long* __restrict__ ei, float* __restrict__ deg,
                          long long E_) {
    long long e = (long long)blockIdx.x * blockDim.x + threadIdx.x;
    if (e < E_) {
        long long d = ei[E_ + e];   // edge_index[1][e]
        atomicAdd(&deg[d], 1.0f);
    }
}

__global__ void deg_rsqrt(float* __restrict__ deg, int n) {
    int i = blockIdx.x * blockDim.x + threadIdx.x;
    if (i < n) deg[i] = rsqrtf(deg[i]);   // deg >= 1 always (self loop)
}

// ---------------------------------------------------------------------------
// 2) h0 = x @ gcn_w via V_WMMA_F32_16X16X4_F32 (FP32 matrix path, wave32)
//    Block: 256 thr = 8 waves -> 32x64 output tile (2 Mtiles x 4 Ntiles).
//    x tile (32x128, 16KB) and full W (128x64, 32KB) staged in LDS with
//    GLOBAL_LOAD_ASYNC_TO_LDS_B128 (ASYNCcnt-tracked direct memory->LDS DMA).
// ---------------------------------------------------------------------------
__global__ __launch_bounds__(256) void gemm_xw(const float* __restrict__ X,
                                               const float* __restrict__ W,
                                               float* __restrict__ H0, int nrows) {
    __shared__ float sX[32 * IN_D];    // 16 KB
    __shared__ float sW[IN_D * HID];   // 32 KB
    const int tid  = threadIdx.x;
    const int lane = tid & 31;
    const int wave = tid >> 5;               // 0..7
    const int row_base = blockIdx.x * 32;

    // LDS aperture flat address keeps the wave-relative byte offset in [31:0]
    const unsigned ldsX = (unsigned)(unsigned long long)(uintptr_t)sX;
    const unsigned ldsW = (unsigned)(unsigned long long)(uintptr_t)sW;

    // async X-tile staging: 32 rows x 128 cols = 1024 x b128, 4 per thread
    #pragma unroll
    for (int i = 0; i < 4; ++i) {
        int i4   = tid + i * 256;            // 0..1023
        int rloc = i4 >> 5;                  // 32 float4 per row
        int c4   = i4 & 31;
        int r    = row_base + rloc;
        if (r >= nrows) r = nrows - 1;       // clamp (N%32==0 in practice)
        unsigned long long ga =
            (unsigned long long)(uintptr_t)(X + (size_t)r * IN_D + (size_t)c4 * 4);
        unsigned la = ldsX + (unsigned)i4 * 16u;
        asm volatile("global_load_async_to_lds_b128 %0, %1, off"
                     :: "v"(la), "v"(ga) : "memory");
    }
    // async W staging: 128x64 = 2048 x b128, 8 per thread
    #pragma unroll
    for (int i = 0; i < 8; ++i) {
        int i4 = tid + i * 256;              // 0..2047
        unsigned long long ga =
            (unsigned long long)(uintptr_t)(W + (size_t)i4 * 4);
        unsigned la = ldsW + (unsigned)i4 * 16u;
        asm volatile("global_load_async_to_lds_b128 %0, %1, off"
                     :: "v"(la), "v"(ga) : "memory");
    }
    asm volatile("s_wait_asynccnt 0" ::: "memory");
    __syncthreads();

    const int mt = (wave & 1) * 16;          // M sub-tile in block tile
    const int nt = (wave >> 1) * 16;         // N sub-tile
    const int mr = lane & 15;                // A row / B-C-D column
    const int kh = (lane >> 4) * 2;          // K half-select per ISA layout

    v8f c = {};
    #pragma unroll
    for (int k = 0; k < IN_D; k += 4) {
        v2f a, b;
        // A 16x4: VGPR0 = K(kh), VGPR1 = K(kh+1), rows on lanes 0..15
        a.x = sX[(mt + mr) * IN_D + k + kh];
        a.y = sX[(mt + mr) * IN_D + k + kh + 1];
        // B 4x16: rows striped across lanes within a VGPR, K halves per lane group
        b.x = sW[(k + kh) * HID + nt + mr];
        b.y = sW[(k + kh + 1) * HID + nt + mr];
        c = __builtin_amdgcn_wmma_f32_16x16x4_f32(false, a, false, b,
                                                  (short)0, c, false, false);
    }

    // D 16x16 f32: VGPR v holds rows v (lanes 0-15) and v+8 (lanes 16-31)
    #pragma unroll
    for (int v = 0; v < 8; ++v) {
        int r = row_base + mt + v + 8 * (lane >> 4);
        if (r < nrows) H0[(size_t)r * HID + nt + (lane & 15)] = c[v];
    }
}

// ---------------------------------------------------------------------------
// 3) Edge messages: one wave per edge, 2 channels per lane, L2-resident atomics
// ---------------------------------------------------------------------------
__global__ __launch_bounds__(256) void edge_msg(const long long* __restrict__ ei,
                                                const float* __restrict__ h0,
                                                const float* __restrict__ dinv,
                                                float* __restrict__ accum,
                                                long long E_) {
    long long w  = ((long long)blockIdx.x * blockDim.x + threadIdx.x) >> 5;
    int lane     = threadIdx.x & 31;
    long long nw = ((long long)gridDim.x * blockDim.x) >> 5;
    for (long long e = w; e < E_; e += nw) {
        long long s = ei[e];
        long long d = ei[E_ + e];
        if (lane == 0 && e + nw < E_) {     // prefetch next src row into cache
            long long s2 = ei[e + nw];
            __builtin_prefetch(h0 + s2 * HID, 0, 1);
        }
        float norm = dinv[s] * dinv[d];
        float2 v = ((const float2*)(h0 + s * HID))[lane];
        float* dst = accum + d * HID + 2 * lane;
        atomicAdd(dst,     v.x * norm);
        atomicAdd(dst + 1, v.y * norm);
    }
}

// ---------------------------------------------------------------------------
// 4) Finalize node features (self-loop + bias + ReLU) and mean-pool atomics
// ---------------------------------------------------------------------------
__global__ void finalize_pool(const float* __restrict__ accum,
                              const float* __restrict__ h0,
                              const float* __restrict__ dinv,
                              const float* __restrict__ gcn_b,
                              const long long* __restrict__ batch,
                              float* __restrict__ sums, float* __restrict__ cnt,
                              long long total) {
    long long idx = (long long)blockIdx.x * blockDim.x + threadIdx.x;
    if (idx >= total) return;
    long long n = idx >> 6;            // / HID
    int c = (int)(idx & 63);
    float di = dinv[n];
    float h = accum[idx] + h0[idx] * di * di + gcn_b[c];
    h = fmaxf(h, 0.0f);
    long long g = batch[n];
    atomicAdd(&sums[g * HID + c], h);
    if (c == 0) atomicAdd(&cnt[g], 1.0f);
}

// ---------------------------------------------------------------------------
// 5) Per-graph heads: doc MLP + layernorm + fusion + task/time
// ---------------------------------------------------------------------------
__global__ __launch_bounds__(128) void heads(const float* __restrict__ sums,
                                             const float* __restrict__ cnt,
                                             const float* __restrict__ docf,
                                             const float* __restrict__ doc_w,
                                             const float* __restrict__ doc_b,
                                             const float* __restrict__ ln_g,
                                             const float* __restrict__ ln_b,
                                             const float* __restrict__ fus_w,
                                             const float* __restrict__ fus_b,
                                             const float* __restrict__ task_w,
                                             const float* __restrict__ task_b,
                                             const float* __restrict__ time_w,
                                             const float* __restrict__ time_b,
                                             float* __restrict__ out_task,
                                             float* __restrict__ out_time) {
    __shared__ float z[2 * HID];
    __shared__ float f[HID];
    __shared__ float mu_s, var_s;
    const int g = blockIdx.x;
    const int t = threadIdx.x;

    if (t < HID) {
        float c = cnt[g];
        if (c < 1.0f) c = 1.0f;
        z[t] = sums[g * HID + t] / c;                     // pooled
    } else {
        int j = t - HID;
        float acc = doc_b[j];
        for (int k = 0; k < DOC_D; ++k)
            acc += docf[g * DOC_D + k] * doc_w[k * HID + j];
        z[HID + j] = fmaxf(acc, 0.0f);                    // doc
    }
    __syncthreads();

    if (t == 0) {
        float s = 0.0f;
        for (int i = 0; i < 2 * HID; ++i) s += z[i];
        float mu = s * (1.0f / (2 * HID));
        float v = 0.0f;
        for (int i = 0; i < 2 * HID; ++i) { float d = z[i] - mu; v += d * d; }
        mu_s = mu; var_s = v * (1.0f / (2 * HID));
    }
    __syncthreads();

    float zi = (z[t] - mu_s) * rsqrtf(var_s + LN_EPS) * ln_g[t] + ln_b[t];
    __syncthreads();
    z[t] = zi;
    __syncthreads();

    if (t < HID) {
        float acc = fus_b[t];
        for (int k = 0; k < 2 * HID; ++k) acc += z[k] * fus_w[k * HID + t];
        f[t] = fmaxf(acc, 0.0f);
    }
    __syncthreads();

    if (t < OUT_D) {
        float acc = task_b[t];
        for (int k = 0; k < HID; ++k) acc += f[k] * task_w[k * OUT_D + t];
        out_task[g * OUT_D + t] = acc;
    }
    if (t == OUT_D) {
        float acc = time_b[0];
        for (int k = 0; k < HID; ++k) acc += f[k] * time_w[k];
        out_time[g] = acc;
    }
}

// ---------------------------------------------------------------------------
extern "C" void kernel_launch(void* const* d_in, const int* in_sizes, int n_in,
                              void* d_out, int out_size, void* d_ws, size_t ws_size,
                              hipStream_t stream) {
    const float*     x      = (const float*)d_in[0];
    const long long* ei     = (const long long*)d_in[1];
    const long long* batch  = (const long long*)d_in[2];
    const float*     docf   = (const float*)d_in[3];
    const float*     gcn_w  = (const float*)d_in[4];
    const float*     gcn_b  = (const float*)d_in[5];
    const float*     doc_w  = (const float*)d_in[6];
    const float*     doc_b  = (const float*)d_in[7];
    const float*     ln_g   = (const float*)d_in[8];
    const float*     ln_b   = (const float*)d_in[9];
    const float*     fus_w  = (const float*)d_in[10];
    const float*     fus_b  = (const float*)d_in[11];
    const float*     task_w = (const float*)d_in[12];
    const float*     task_b = (const float*)d_in[13];
    const float*     time_w = (const float*)d_in[14];
    const float*     time_b = (const float*)d_in[15];

    const int       N_ = in_sizes[0] / IN_D;
    const long long E_ = (long long)in_sizes[1] / 2;
    const int       G_ = in_sizes[3] / DOC_D;

    // workspace layout (256B aligned)
    char* ws = (char*)d_ws;
    size_t off = 0;
    auto alignup = [](size_t v) { return (v + 255) & ~(size_t)255; };
    float* deg   = (float*)(ws + off); off = alignup(off + (size_t)N_ * 4);
    float* h0    = (float*)(ws + off); off = alignup(off + (size_t)N_ * HID * 4);
    float* accum = (float*)(ws + off); off = alignup(off + (size_t)N_ * HID * 4);
    float* sums  = (float*)(ws + off); off = alignup(off + (size_t)G_ * HID * 4);
    float* cnt   = (float*)(ws + off); off = alignup(off + (size_t)G_ * 4);
    (void)ws_size;

    float* out_task = (float*)d_out;                  // [G, OUT_D]
    float* out_time = out_task + (size_t)G_ * OUT_D;  // [G, 1]

    const long long accumN = (long long)N_ * HID;

    init_ws<<<2048, 256, 0, stream>>>(deg, accum, sums, cnt, N_, accumN, G_);

    {
        int blk = (int)((E_ + 255) / 256);
        deg_accum<<<blk, 256, 0, stream>>>(ei, deg, E_);
    }
    deg_rsqrt<<<(N_ + 255) / 256, 256, 0, stream>>>(deg, N_);

    gemm_xw<<<(N_ + 31) / 32, 256, 0, stream>>>(x, gcn_w, h0, N_);

    edge_msg<<<2048, 256, 0, stream>>>(ei, h0, deg, accum, E_);

    {
        long long total = accumN;
        int blk = (int)((total + 255) / 256);
        finalize_pool<<<blk, 256, 0, stream>>>(accum, h0, deg, gcn_b, batch,
                                               sums, cnt, total);
    }

    heads<<<G_, 128, 0, stream>>>(sums, cnt, docf, doc_w, doc_b, ln_g, ln_b,
                                  fus_w, fus_b, task_w, task_b, time_w, time_b,
                                  out_task, out_time);
}